// DetectionLoss_28441273434222
// MI455X (gfx1250) — compile-verified
//
#include <hip/hip_runtime.h>
#include <hip/hip_bf16.h>

// ---------------- problem constants (match reference) ----------------
#define BATCH 8
#define A_ANCH 9
#define NC 90
#define NTOT 49104            // A * sum(hw^2) over 5 levels
#define BOX_W 50.0f
#define EPSV 1e-7f

typedef __attribute__((ext_vector_type(2))) float v2f;
typedef __attribute__((ext_vector_type(8))) float v8f;

// ---------------------------------------------------------------------
// Class focal-loss partial reduction.
// One launch per pyramid level. `e` walks the *native* layout of
// cls_out [B, A*NC, H, W] so the big 141MB stream is fully coalesced.
// ---------------------------------------------------------------------
__global__ void __launch_bounds__(256)
cls_focal_partial(const float* __restrict__ logits,
                  const int*   __restrict__ tgt,
                  float* __restrict__ partial,
                  int H, int E) {
  const int W = H;
  float acc = 0.0f;
  const int stride = gridDim.x * blockDim.x;
  for (int e = blockIdx.x * blockDim.x + threadIdx.x; e < E; e += stride) {
    int w  = e % W;            int t1 = e / W;
    int h  = t1 % H;           t1 /= H;
    int ch = t1 % (A_ANCH*NC); int b  = t1 / (A_ANCH*NC);
    int a  = ch / NC;          int c  = ch - a*NC;

    int ct = tgt[((b*H + h)*W + w)*A_ANCH + a];
    if (ct != -2) {
      float x   = logits[e];                    // e IS the linear index
      float isT = (ct == c) ? 1.0f : 0.0f;      // ct==c implies ct>=0 (valid)
      float p   = 1.0f / (1.0f + __expf(-x));
      float pt  = isT * p + (1.0f - isT) * (1.0f - p);
      float af  = isT * 0.25f + (1.0f - isT) * 0.75f;
      float u   = 1.0f - pt;
      float mod = u * __fsqrt_rn(u);            // u^1.5, u >= 0
      float bce = fmaxf(x, 0.0f) - x*isT + __logf(1.0f + __expf(-fabsf(x)));
      acc += af * mod * bce;
    }
  }
  __shared__ float red[256];
  red[threadIdx.x] = acc;
  __syncthreads();
  for (int off = 128; off > 0; off >>= 1) {
    if (threadIdx.x < off) red[threadIdx.x] += red[threadIdx.x + off];
    __syncthreads();
  }
  if (threadIdx.x == 0) partial[blockIdx.x] = red[0];
}

// ---------------------------------------------------------------------
// Box GIoU partial reduction over all B*NTOT anchors (all 5 levels).
// ---------------------------------------------------------------------
__global__ void __launch_bounds__(256)
box_giou_partial(const float* __restrict__ bo0, const float* __restrict__ bo1,
                 const float* __restrict__ bo2, const float* __restrict__ bo3,
                 const float* __restrict__ bo4,
                 const float* __restrict__ bt0, const float* __restrict__ bt1,
                 const float* __restrict__ bt2, const float* __restrict__ bt3,
                 const float* __restrict__ bt4,
                 const float* __restrict__ anch,
                 float* __restrict__ pa_part, float* __restrict__ mk_part,
                 int total) {
  float accP = 0.0f, accM = 0.0f;
  const int stride = gridDim.x * blockDim.x;
  for (int idx = blockIdx.x * blockDim.x + threadIdx.x; idx < total; idx += stride) {
    int n = idx % NTOT;
    int b = idx / NTOT;

    int H, base; const float *bo, *bt;
    if      (n < 36864) { H = 64; base = 0;     bo = bo0; bt = bt0; }
    else if (n < 46080) { H = 32; base = 36864; bo = bo1; bt = bt1; }
    else if (n < 48384) { H = 16; base = 46080; bo = bo2; bt = bt2; }
    else if (n < 48960) { H = 8;  base = 48384; bo = bo3; bt = bt3; }
    else                { H = 4;  base = 48960; bo = bo4; bt = bt4; }

    int nl  = n - base;
    int a   = nl % A_ANCH;
    int pix = nl / A_ANCH;
    int w   = pix % H;
    int h   = pix / H;

    // targets: [B,H,W,A*4]
    int ti = ((b*H + h)*H + w)*(A_ANCH*4) + a*4;
    float ty_t = bt[ti+0], tx_t = bt[ti+1], th_t = bt[ti+2], tw_t = bt[ti+3];

    bool pos = (ty_t != 0.0f) && (tx_t != 0.0f) && (th_t != 0.0f) && (tw_t != 0.0f);
    if (pos) {
      accM += 1.0f;
      // outputs: [B, A*4, H, W]
      int oi0 = ((b*(A_ANCH*4) + a*4 + 0)*H + h)*H + w;
      int HW  = H*H;
      float ty_o = bo[oi0], tx_o = bo[oi0+HW], th_o = bo[oi0+2*HW], tw_o = bo[oi0+3*HW];

      float ay1 = anch[4*n+0], ax1 = anch[4*n+1], ay2 = anch[4*n+2], ax2 = anch[4*n+3];
      float ha  = ay2 - ay1,   wa  = ax2 - ax1;
      float yca = 0.5f*(ay1+ay2), xca = 0.5f*(ax1+ax2);

      // decode target
      float hh = __expf(th_t) * ha, wwv = __expf(tw_t) * wa;
      float yc = ty_t * ha + yca,   xc  = tx_t * wa + xca;
      float dt0 = yc - 0.5f*hh, dt1 = xc - 0.5f*wwv, dt2 = yc + 0.5f*hh, dt3 = xc + 0.5f*wwv;
      // decode output
      float ho = __expf(th_o) * ha, wo = __expf(tw_o) * wa;
      float yo = ty_o * ha + yca,   xo = tx_o * wa + xca;
      float do0 = yo - 0.5f*ho, do1 = xo - 0.5f*wo, do2 = yo + 0.5f*ho, do3 = xo + 0.5f*wo;

      float Ag = (dt3 - dt1) * (dt2 - dt0);
      float Ap = (do3 - do1) * (do2 - do0);
      float yi1 = fmaxf(dt0, do0), xi1 = fmaxf(dt1, do1);
      float yi2 = fminf(dt2, do2), xi2 = fminf(dt3, do3);
      float I  = ((xi2 > xi1) && (yi2 > yi1)) ? (xi2 - xi1)*(yi2 - yi1) : 0.0f;
      float U  = Ap + Ag - I;
      float iou = I / (U + EPSV);
      float yc1 = fminf(dt0, do0), xc1 = fminf(dt1, do1);
      float yc2 = fmaxf(dt2, do2), xc2 = fmaxf(dt3, do3);
      float Ac  = (xc2 - xc1) * (yc2 - yc1);
      float pen = (Ac - U) / (Ac + EPSV);
      accP += (1.0f - iou + pen);
    }
  }
  __shared__ float redP[256];
  __shared__ float redM[256];
  redP[threadIdx.x] = accP;
  redM[threadIdx.x] = accM;
  __syncthreads();
  for (int off = 128; off > 0; off >>= 1) {
    if (threadIdx.x < off) {
      redP[threadIdx.x] += redP[threadIdx.x + off];
      redM[threadIdx.x] += redM[threadIdx.x + off];
    }
    __syncthreads();
  }
  if (threadIdx.x == 0) {
    pa_part[blockIdx.x] = redP[0];
    mk_part[blockIdx.x] = redM[0];
  }
}

// ---------------------------------------------------------------------
// Deterministic WMMA reduction of n floats (n must be a multiple of 64).
// A = 16x4 chunk of partials, B = ones(4x16)  ->  D[m,n] = rowsum(m)+C.
// ISA A layout: lane L holds (m=L%16, k = 2*(L/16) .. +1).
// C/D layout: lanes 0-15 VGPR r -> M=r ; lanes 16-31 -> M=r+8.
// Requires a single fully-active wave32 (EXEC all ones).
// ---------------------------------------------------------------------
__device__ __forceinline__ float wmma_reduce64(const float* __restrict__ p, int n) {
  const int lane = threadIdx.x & 31;
  const int m  = lane & 15;
  const int kb = (lane >> 4) << 1;
  v2f ones; ones.x = 1.0f; ones.y = 1.0f;
  v8f c = {};
  for (int base = 0; base < n; base += 64) {
    v2f a;
    int i0 = base + m*4 + kb;
    a.x = p[i0];
    a.y = p[i0 + 1];
    // D = A(16x4) * ones(4x16) + C  -> each D[m,*] = rowsum(m) accumulated
    c = __builtin_amdgcn_wmma_f32_16x16x4_f32(false, a, false, ones,
                                              (short)0, c, false, false);
  }
  float s = c[0]+c[1]+c[2]+c[3]+c[4]+c[5]+c[6]+c[7]; // sum over 8 M-rows in this half
  s += __shfl_xor(s, 16, 32);                        // low half (m 0-7) + high half (m 8-15)
  return s;                                          // grand total, in every lane
}

__global__ void finalize_loss(const float* __restrict__ ws_cls, int n_cls,
                              const float* __restrict__ ws_pa,
                              const float* __restrict__ ws_mk, int n_box,
                              const float* __restrict__ num_pos,
                              float* __restrict__ out) {
  float cls_sum = wmma_reduce64(ws_cls, n_cls);
  float pa_sum  = wmma_reduce64(ws_pa,  n_box);
  float mk_sum  = wmma_reduce64(ws_mk,  n_box);

  float nps = 1.0f;
#pragma unroll
  for (int i = 0; i < BATCH; ++i) nps += num_pos[i];

  float cls_loss = cls_sum / nps;
  float box_loss = pa_sum / mk_sum;
  if (threadIdx.x == 0) {
    out[0] = cls_loss + BOX_W * box_loss;
    out[1] = cls_loss;
    out[2] = box_loss;
  }
}

// ---------------------------------------------------------------------
// kernel_launch
// d_in dict order per level i: 4i cls_out, 4i+1 box_out, 4i+2 cls_tgt,
// 4i+3 box_tgt; then [20] num_positives, [21] anchors.
// ---------------------------------------------------------------------
extern "C" void kernel_launch(void* const* d_in, const int* in_sizes, int n_in,
                              void* d_out, int out_size, void* d_ws, size_t ws_size,
                              hipStream_t stream) {
  const float* cls_out[5]; const float* box_out[5];
  const int*   cls_tgt[5]; const float* box_tgt[5];
  for (int i = 0; i < 5; ++i) {
    cls_out[i] = (const float*)d_in[4*i + 0];
    box_out[i] = (const float*)d_in[4*i + 1];
    cls_tgt[i] = (const int*)  d_in[4*i + 2];
    box_tgt[i] = (const float*)d_in[4*i + 3];
  }
  const float* num_pos = (const float*)d_in[20];
  const float* anchors = (const float*)d_in[21];
  float* out = (float*)d_out;

  // workspace: cls partials (1856 = 29*64) | box pa (512) | box mask (512)
  float* ws_cls = (float*)d_ws;
  float* ws_pa  = ws_cls + 1856;
  float* ws_mk  = ws_pa  + 512;

  const int HWs[5]    = {64, 32, 16, 8, 4};
  const int blocks[5] = {896, 512, 256, 128, 64};  // sum = 1856, multiple of 64

  int off = 0;
  for (int i = 0; i < 5; ++i) {
    int H = HWs[i];
    int E = BATCH * A_ANCH * NC * H * H;
    cls_focal_partial<<<blocks[i], 256, 0, stream>>>(cls_out[i], cls_tgt[i],
                                                     ws_cls + off, H, E);
    off += blocks[i];
  }

  box_giou_partial<<<512, 256, 0, stream>>>(
      box_out[0], box_out[1], box_out[2], box_out[3], box_out[4],
      box_tgt[0], box_tgt[1], box_tgt[2], box_tgt[3], box_tgt[4],
      anchors, ws_pa, ws_mk, BATCH * NTOT);

  // one full wave32, all lanes active (WMMA EXEC requirement)
  finalize_loss<<<1, 32, 0, stream>>>(ws_cls, 1856, ws_pa, ws_mk, 512,
                                      num_pos, out);
}